// SpatialSmoothnessLoss_41042707481152
// MI455X (gfx1250) — compile-verified
//
#include <hip/hip_runtime.h>
#include <math.h>

#define ALPHA_INV 2.0f      // 1/0.5
#define NUM_NB 8
#define BATCH 4
#define NPTS 8192
#define QTILE 16
#define WAVES_PER_BLOCK 8
#define THREADS (WAVES_PER_BLOCK * 32)

typedef __attribute__((ext_vector_type(2))) float v2f;
typedef __attribute__((ext_vector_type(8))) float v8f;

__global__ void ws_zero_kernel(float* ws) {
    if (threadIdx.x < 2 * BATCH) ws[threadIdx.x] = 0.0f;
}

__launch_bounds__(THREADS)
__global__ void knn_loss_kernel(const float* __restrict__ pc1,
                                const float* __restrict__ flow,
                                float* __restrict__ ws) {
    // Dynamic LDS: x / y / z / |p|^2 for the whole batch (4 * 8192 * 4B = 128 KB)
    extern __shared__ float smem[];
    float* sx  = smem;
    float* sy  = smem + NPTS;
    float* sz  = smem + 2 * NPTS;
    float* ssq = smem + 3 * NPTS;

    const int wave = threadIdx.x >> 5;
    const int lane = threadIdx.x & 31;
    const int blocksPerBatch = (NPTS / QTILE) / WAVES_PER_BLOCK; // 64
    const int b    = blockIdx.x / blocksPerBatch;
    const int tile = (blockIdx.x % blocksPerBatch) * WAVES_PER_BLOCK + wave;
    const int q0   = tile * QTILE;

    const float* xc = pc1  + (size_t)(b * 3 + 0) * NPTS;
    const float* yc = pc1  + (size_t)(b * 3 + 1) * NPTS;
    const float* zc = pc1  + (size_t)(b * 3 + 2) * NPTS;
    const float* fx = flow + (size_t)(b * 3 + 0) * NPTS;
    const float* fy = flow + (size_t)(b * 3 + 1) * NPTS;
    const float* fz = flow + (size_t)(b * 3 + 2) * NPTS;

    // ---- cooperative stage of candidate data into LDS (coalesced, once) ----
    for (int i = threadIdx.x; i < NPTS; i += THREADS) {
        float x = xc[i], y = yc[i], z = zc[i];
        sx[i] = x; sy[i] = y; sz[i] = z;
        ssq[i] = x * x + y * y + z * z;
    }
    __syncthreads();

    const int  m   = lane & 15;
    const bool hi  = lane >= 16;
    const int  myq = q0 + m;

    // warm flow rows for the tail gathers
    __builtin_prefetch(&fx[myq], 0, 1);
    __builtin_prefetch(&fy[myq], 0, 1);
    __builtin_prefetch(&fz[myq], 0, 1);

    // ---- B fragment: fixed query tile, col n = m ----
    // lanes 0-15 : (K0,K1) = (-2x, -2y);  lanes 16-31 : (K2,K3) = (-2z, 1)
    v2f bfrag;
    bfrag.x = -2.0f * (hi ? sz[myq] : sx[myq]);
    bfrag.y = hi ? 1.0f : (-2.0f * sy[myq]);
    const float qsq = ssq[myq];                 // |p_q|^2, per-lane scalar

    // A-fragment source pointers for this lane half
    const float* pa0 = hi ? sz  : sx;
    const float* pa1 = hi ? ssq : sy;

    // ---- unsorted top-8 with tracked max slot (branchless replace) ----
    float dl[NUM_NB]; int il[NUM_NB];
#pragma unroll
    for (int k = 0; k < NUM_NB; ++k) { dl[k] = 3.4e38f; il[k] = -1; }
    float dmax = 3.4e38f;
    int   slot = 0;

    auto insertCand = [&](float d, int ci) {
#pragma unroll
        for (int j = 0; j < NUM_NB; ++j) {
            bool w = (j == slot);
            dl[j] = w ? d  : dl[j];
            il[j] = w ? ci : il[j];
        }
        dmax = dl[0]; slot = 0;
#pragma unroll
        for (int j = 1; j < NUM_NB; ++j) {
            bool g = dl[j] > dmax;
            dmax = g ? dl[j] : dmax;
            slot = g ? j : slot;
        }
    };

    // ---- stream candidate tiles (A matrix) from LDS, software-pipelined ----
    // Prologue: loads for tile 0 in flight before the loop.
    v2f afrag;
    afrag.x = pa0[m];
    afrag.y = pa1[m];

#pragma unroll 2
    for (int base = 0; base < NPTS; base += QTILE) {
        const v2f cur = afrag;

        // Issue next tile's LDS loads early so s_wait_dscnt before the next
        // WMMA can overlap with this tile's selection work. Branchless wrap:
        // last iteration redundantly reloads tile 0 (result unused).
        const int nc = ((base + QTILE) & (NPTS - 1)) + m;
        afrag.x = pa0[nc];               // x (lo) | z (hi)
        afrag.y = pa1[nc];               // y (lo) | |p_c|^2 (hi)

        v8f acc = {};
        // C[m][n] = |p_m|^2 - 2 * p_m . p_n   (m = candidate row, n = query col)
        acc = __builtin_amdgcn_wmma_f32_16x16x4_f32(
            /*neg_a=*/false, cur, /*neg_b=*/false, bfrag,
            /*c_mod=*/(short)0, acc, /*reuse_a=*/false, /*reuse_b=*/false);

        const int mbase = base + (hi ? 8 : 0);
#pragma unroll
        for (int v = 0; v < 8; ++v) {
            float d  = acc[v] + qsq;     // full squared distance
            int   ci = mbase + v;
            if (ci != myq && d < dmax) insertCand(d, ci);
        }
    }

    // ---- merge lane pairs (L, L+16): snapshot partner list, then insert ----
    float od[NUM_NB]; int oi[NUM_NB];
#pragma unroll
    for (int k = 0; k < NUM_NB; ++k) {
        od[k] = __shfl_xor(dl[k], 16);
        oi[k] = __shfl_xor(il[k], 16);
    }
#pragma unroll
    for (int k = 0; k < NUM_NB; ++k) {
        if (od[k] < dmax) insertCand(od[k], oi[k]);
    }

    // ---- per-query contributions (valid on lanes 0-15) ----
    float esum = 0.0f, usum = 0.0f;
    if (!hi) {
        float fqx = fx[myq], fqy = fy[myq], fqz = fz[myq];
#pragma unroll
        for (int k = 0; k < NUM_NB; ++k) {
            int   j  = il[k];
            float dd = fmaxf(dl[k], 0.0f);
            float zz = __expf(-dd * ALPHA_INV);   // exp(-d/alpha), in (0,1]
            float e  = __expf(zz);                // softmax numerator, safe range
            float gx = fx[j] - fqx;
            float gy = fy[j] - fqy;
            float gz = fz[j] - fqz;
            float diff = sqrtf(gx * gx + gy * gy + gz * gz);
            esum += e;
            usum += e * diff;
        }
    }

    // ---- wave reduction then per-batch atomic accumulation ----
#pragma unroll
    for (int off = 16; off >= 1; off >>= 1) {
        esum += __shfl_xor(esum, off);
        usum += __shfl_xor(usum, off);
    }
    if (lane == 0) {
        atomicAdd(&ws[2 * b + 0], esum);
        atomicAdd(&ws[2 * b + 1], usum);
    }
}

__global__ void finalize_kernel(const float* __restrict__ ws, float* __restrict__ out) {
    if (threadIdx.x == 0) {
        float acc = 0.0f;
#pragma unroll
        for (int b = 0; b < BATCH; ++b) acc += ws[2 * b + 1] / ws[2 * b + 0];
        out[0] = acc / (float)BATCH;
    }
}

extern "C" void kernel_launch(void* const* d_in, const int* in_sizes, int n_in,
                              void* d_out, int out_size, void* d_ws, size_t ws_size,
                              hipStream_t stream) {
    (void)in_sizes; (void)n_in; (void)out_size; (void)ws_size;
    const float* pc1  = (const float*)d_in[0];
    const float* flow = (const float*)d_in[1];
    float* ws  = (float*)d_ws;
    float* out = (float*)d_out;

    ws_zero_kernel<<<1, 32, 0, stream>>>(ws);

    const int blocksPerBatch = (NPTS / QTILE) / WAVES_PER_BLOCK; // 64
    dim3 grid(BATCH * blocksPerBatch);
    dim3 block(THREADS);
    size_t ldsBytes = (size_t)4 * NPTS * sizeof(float);          // 128 KB
    knn_loss_kernel<<<grid, block, ldsBytes, stream>>>(pc1, flow, ws);

    finalize_kernel<<<1, 32, 0, stream>>>(ws, out);
}